// Int2TinyLSTM_86560770884278
// MI455X (gfx1250) — compile-verified
//
#include <hip/hip_runtime.h>
#include <hip/hip_bf16.h>
#include <math.h>

typedef __attribute__((ext_vector_type(2))) float v2f;
typedef __attribute__((ext_vector_type(8))) float v8f;

#define V_SZ 32000
#define D_SZ 32
#define B_SZ 8
#define S_SZ 1024
#define G_SZ 128              // 4*D
#define NTOK (B_SZ * S_SZ)    // 8192

__device__ __forceinline__ float quantw(float w, float s, float r) {
    // q = clip(round(w/s*2)/2, -1.5, 1.5) * s ;  r = 2/s ; rintf == RNE
    float t = rintf(w * r) * 0.5f;
    t = fminf(1.5f, fmaxf(-1.5f, t));
    return t * s;
}

__device__ __forceinline__ float sigm(float x) { return 1.0f / (1.0f + __expf(-x)); }

// ---------------- stage 1: per-block partial sums of |w| (deterministic) ---------
__global__ void absum_partial(const float* __restrict__ in, int n, float* __restrict__ partials) {
    __shared__ float red[256];
    float s = 0.0f;
    for (int i = blockIdx.x * blockDim.x + threadIdx.x; i < n; i += gridDim.x * blockDim.x)
        s += fabsf(in[i]);
    red[threadIdx.x] = s;
    __syncthreads();
    for (int off = 128; off > 0; off >>= 1) {
        if (threadIdx.x < off) red[threadIdx.x] += red[threadIdx.x + off];
        __syncthreads();
    }
    if (threadIdx.x == 0) partials[blockIdx.x] = red[0];
}

// ---------------- stage 2: 4 scales = max(mean*2, 1e-8) --------------------------
__global__ void finalize_scales(float* __restrict__ ws) {
    __shared__ float red[128];
    const float ns[4] = { (float)(V_SZ * D_SZ), (float)(G_SZ * D_SZ),
                          (float)(G_SZ * D_SZ), (float)(V_SZ * D_SZ) };
    for (int t = 0; t < 4; ++t) {
        red[threadIdx.x] = ws[16 + t * 128 + threadIdx.x];
        __syncthreads();
        for (int off = 64; off > 0; off >>= 1) {
            if (threadIdx.x < off) red[threadIdx.x] += red[threadIdx.x + off];
            __syncthreads();
        }
        if (threadIdx.x == 0) ws[t] = fmaxf((red[0] / ns[t]) * 2.0f, 1e-8f);
        __syncthreads();
    }
}

// ---------------- precompute quantized fc_w (removes quant VALU from GEMM) -------
__global__ void quant_fcw_kernel(const float* __restrict__ fc_w, const float* __restrict__ ws,
                                 float* __restrict__ fcq) {
    const float s = ws[3], r = 2.0f / s;
    const int i = blockIdx.x * blockDim.x + threadIdx.x;   // exactly V*D threads
    fcq[i] = quantw(fc_w[i], s, r);
}

// ---------------- x_gates[b,s,g] = q(emb)[x[b,s]] . q(Wih[g]) + b_ih[g] ----------
__global__ void xgates_kernel(const int* __restrict__ x, const float* __restrict__ emb_w,
                              const float* __restrict__ w_ih, const float* __restrict__ b_ih,
                              const float* __restrict__ ws, float* __restrict__ xg) {
    __shared__ float e[D_SZ];
    const int t = blockIdx.x;                 // token 0..8191 (flat b*S+s)
    const float s_emb = ws[0], r_emb = 2.0f / s_emb;
    const float s_ih = ws[1], r_ih = 2.0f / s_ih;
    const int tok = x[t];
    if (threadIdx.x < D_SZ)
        e[threadIdx.x] = quantw(emb_w[(size_t)tok * D_SZ + threadIdx.x], s_emb, r_emb);
    __syncthreads();
    const int g = threadIdx.x;                // 0..127
    float acc = b_ih[g];
#pragma unroll
    for (int d = 0; d < D_SZ; ++d)
        acc += e[d] * quantw(w_ih[g * D_SZ + d], s_ih, r_ih);
    xg[(size_t)t * G_SZ + g] = acc;
}

// ---------------- sequential LSTM scan: one workgroup, 1024 threads --------------
__global__ void lstm_kernel(const float* __restrict__ xg, const float* __restrict__ w_hh,
                            const float* __restrict__ b_hh, const float* __restrict__ ws,
                            float* __restrict__ outs, float* __restrict__ tail) {
    __shared__ __align__(16) float whh[G_SZ * D_SZ];   // quantized Whh, 16 KB
    __shared__ float bh[G_SZ];
    __shared__ __align__(16) float h[B_SZ * D_SZ];
    __shared__ float c[B_SZ * D_SZ];
    __shared__ float gates[B_SZ * G_SZ];
    const int tid = threadIdx.x;              // 0..1023
    const float s_hh = ws[2], r_hh = 2.0f / s_hh;
    for (int i = tid; i < G_SZ * D_SZ; i += 1024) whh[i] = quantw(w_hh[i], s_hh, r_hh);
    if (tid < G_SZ) bh[tid] = b_hh[tid];
    if (tid < B_SZ * D_SZ) { h[tid] = 0.0f; c[tid] = 0.0f; }
    __syncthreads();

    const int b = tid >> 7;                   // batch 0..7
    const int g = tid & 127;                  // gate 0..127
    const float4* whh4 = (const float4*)&whh[g * D_SZ];
    const float4* h4   = (const float4*)&h[b * D_SZ];

    for (int s = 0; s < S_SZ; ++s) {
        float acc = xg[((size_t)b * S_SZ + s) * G_SZ + g] + bh[g];
#pragma unroll
        for (int d4 = 0; d4 < D_SZ / 4; ++d4) {
            float4 hv = h4[d4];
            float4 wv = whh4[d4];
            acc += hv.x * wv.x + hv.y * wv.y + hv.z * wv.z + hv.w * wv.w;
        }
        gates[tid] = acc;
        __syncthreads();
        if (tid < B_SZ * D_SZ) {
            const int bb = tid >> 5, d = tid & 31;
            const float ig = gates[bb * G_SZ + d];
            const float fg = gates[bb * G_SZ + 32 + d];
            const float gg = gates[bb * G_SZ + 64 + d];
            const float og = gates[bb * G_SZ + 96 + d];
            const float cn = sigm(fg) * c[tid] + sigm(ig) * tanhf(gg);
            const float hn = sigm(og) * tanhf(cn);
            c[tid] = cn; h[tid] = hn;
            outs[((size_t)bb * S_SZ + s) * D_SZ + d] = hn;
        }
        __syncthreads();
    }
    if (tid < B_SZ * D_SZ) { tail[tid] = h[tid]; tail[B_SZ * D_SZ + tid] = c[tid]; }
}

// ---------------- logits = outs[8192,32] @ fcq^T[32,32000] + fc_b ----------------
// V_WMMA_F32_16X16X4_F32, 4 M-tiles x 4 N-tiles per wave (128 WMMAs), processed as
// 2 M-tile pairs per N-tile so only TWO accumulator chains (16 VGPRs) are live:
//   live set ~ A(64) + B(16) + acc(16) + addr  -> no scratch spills.
// A held across all N-tiles; B loaded once per N-tile; bias preloaded into acc;
// non-temporal stores for the 1.05 GB output.
__global__ void __launch_bounds__(256)
logits_kernel(const float* __restrict__ outs, const float* __restrict__ fcq,
              const float* __restrict__ fc_b, float* __restrict__ out) {
    const int wave  = blockIdx.x * 8 + (threadIdx.x >> 5);
    const int lane  = threadIdx.x & 31;
    const int half  = lane >> 4;              // 0: K,K+1   1: K+2,K+3
    const int l15   = lane & 15;
    const int mgroup = wave & 127;            // 128 groups of 4 M-tiles (8192/64)
    const int ngroup = wave >> 7;             // 500 groups of 4 N-tiles (32000/64)

    // A: 4 tiles of 16 rows x K=32  ->  4 x 8 x v2f per lane (64 VGPRs)
    v2f a[4][8];
#pragma unroll
    for (int m = 0; m < 4; ++m) {
        const int row = (mgroup * 4 + m) * 16 + l15;
        const float* arow = outs + (size_t)row * D_SZ + half * 2;
#pragma unroll
        for (int kk = 0; kk < 8; ++kk)
            a[m][kk] = *(const v2f*)(arow + kk * 4);
    }

#pragma unroll
    for (int nt = 0; nt < 4; ++nt) {
        const int colbase = (ngroup * 4 + nt) * 16;
        const float bias = fc_b[colbase + l15];

        // B tile: B[k][n] = fcq[n][k]  ->  8 x v2f per lane (16 VGPRs)
        const float* brow = fcq + (size_t)(colbase + l15) * D_SZ + half * 2;
        v2f bv[8];
#pragma unroll
        for (int kk = 0; kk < 8; ++kk)
            bv[kk] = *(const v2f*)(brow + kk * 4);

#pragma unroll
        for (int mp = 0; mp < 4; mp += 2) {
            v8f acc0, acc1;
#pragma unroll
            for (int j = 0; j < 8; ++j) { acc0[j] = bias; acc1[j] = bias; }

#pragma unroll
            for (int kk = 0; kk < 8; ++kk) {
                acc0 = __builtin_amdgcn_wmma_f32_16x16x4_f32(false, a[mp][kk],     false, bv[kk],
                                                             (short)0, acc0, false, false);
                acc1 = __builtin_amdgcn_wmma_f32_16x16x4_f32(false, a[mp + 1][kk], false, bv[kk],
                                                             (short)0, acc1, false, false);
            }

            // C VGPR j -> rows j (lanes 0-15) / j+8 (lanes 16-31)
            float* p0 = out + (size_t)((mgroup * 4 + mp)     * 16 + 8 * half) * V_SZ + colbase + l15;
            float* p1 = out + (size_t)((mgroup * 4 + mp + 1) * 16 + 8 * half) * V_SZ + colbase + l15;
#pragma unroll
            for (int j = 0; j < 8; ++j) {
                __builtin_nontemporal_store(acc0[j], p0 + (size_t)j * V_SZ);
                __builtin_nontemporal_store(acc1[j], p1 + (size_t)j * V_SZ);
            }
        }
    }
}

extern "C" void kernel_launch(void* const* d_in, const int* in_sizes, int n_in,
                              void* d_out, int out_size, void* d_ws, size_t ws_size,
                              hipStream_t stream) {
    const int*   x     = (const int*)d_in[0];
    const float* emb_w = (const float*)d_in[1];
    const float* w_ih  = (const float*)d_in[2];
    const float* b_ih  = (const float*)d_in[3];
    const float* w_hh  = (const float*)d_in[4];
    const float* b_hh  = (const float*)d_in[5];
    const float* fc_w  = (const float*)d_in[6];
    const float* fc_b  = (const float*)d_in[7];
    float* out = (float*)d_out;
    float* ws  = (float*)d_ws;

    // ws layout (floats): [0..3] scales, [16..527] partials,
    // [1024..) x_gates (8192*128), outs (8192*32), fcq (32000*32)  ~9.4 MB total
    float* xg    = ws + 1024;
    float* houts = xg + (size_t)NTOK * G_SZ;
    float* fcq   = houts + (size_t)NTOK * D_SZ;

    absum_partial<<<128, 256, 0, stream>>>(emb_w, V_SZ * D_SZ, ws + 16);
    absum_partial<<<128, 256, 0, stream>>>(w_ih,  G_SZ * D_SZ, ws + 16 + 128);
    absum_partial<<<128, 256, 0, stream>>>(w_hh,  G_SZ * D_SZ, ws + 16 + 256);
    absum_partial<<<128, 256, 0, stream>>>(fc_w,  V_SZ * D_SZ, ws + 16 + 384);
    finalize_scales<<<1, 128, 0, stream>>>(ws);

    quant_fcw_kernel<<<(V_SZ * D_SZ) / 256, 256, 0, stream>>>(fc_w, ws, fcq);

    xgates_kernel<<<NTOK, G_SZ, 0, stream>>>(x, emb_w, w_ih, b_ih, ws, xg);

    // logits occupy NTOK*V floats; hT then cT appended
    lstm_kernel<<<1, 1024, 0, stream>>>(xg, w_hh, b_hh, ws, houts,
                                        out + (size_t)NTOK * V_SZ);

    // 128 M-groups * 500 N-groups = 64000 waves / 8 per block
    logits_kernel<<<8000, 256, 0, stream>>>(houts, fcq, fc_b, out);
}